// GCN_62113817035242
// MI455X (gfx1250) — compile-verified
//
#include <hip/hip_runtime.h>

// ---------------------------------------------------------------------------
// 2-layer GCN (PyG GCNConv semantics) for MI455X / gfx1250.
//   layer: out = D^-1/2 (A+I) D^-1/2 (X W) + b
// GEMMs use V_WMMA_F32_16X16X4_F32 (fp32-exact). Aggregation uses native
// fp32 global atomics (memory-bound part: ~1.2 GB @ 23.3 TB/s).
// ---------------------------------------------------------------------------

typedef float v2f __attribute__((ext_vector_type(2)));
typedef float v8f __attribute__((ext_vector_type(8)));

__device__ __forceinline__ void atomic_add_f32(float* p, float v) {
    __hip_atomic_fetch_add(p, v, __ATOMIC_RELAXED, __HIP_MEMORY_SCOPE_AGENT);
}

// ---------------- degree / norm ----------------

__global__ void k_deg_init(unsigned* __restrict__ deg, int n) {
    int i = blockIdx.x * blockDim.x + threadIdx.x;
    if (i < n) deg[i] = 1u;  // self loop
}

__global__ void k_deg_count(const int* __restrict__ dst, unsigned* __restrict__ deg, int e) {
    int i = blockIdx.x * blockDim.x + threadIdx.x;
    if (i < e) atomicAdd(&deg[dst[i]], 1u);
}

__global__ void k_dinv(const unsigned* __restrict__ deg, float* __restrict__ dinv, int n) {
    int i = blockIdx.x * blockDim.x + threadIdx.x;
    if (i < n) dinv[i] = rsqrtf((float)deg[i]);
}

// ---------------- GEMM: D[M,N] = (relu?)A[M,K] * B[K,N] ----------------
// One wave per 16x16 output tile; wave w in the block owns column tile w.
// gridDim.x = M/16 (full tiles), blockDim.x = 32 * (N/16). K % 4 == 0.
//
// V_WMMA_F32_16X16X4_F32 operand layout (ISA 7.12.2, wave32):
//   A (16x4): lane l<16  -> row l,    v0=K0, v1=K1 ; lane l>=16 -> row l-16, v0=K2, v1=K3
//   B (4x16): lane l<16  -> col l,    v0=K0, v1=K1 ; lane l>=16 -> col l-16, v0=K2, v1=K3
//   C/D (16x16): VGPR r : lanes 0-15 -> (M=r,   N=l) ; lanes 16-31 -> (M=r+8, N=l)
template <bool RELU>
__global__ void gemm16_wmma(const float* __restrict__ A, const float* __restrict__ B,
                            float* __restrict__ D, int K, int N) {
    const int lane = threadIdx.x & 31;
    const int wave = threadIdx.x >> 5;
    const int l    = lane & 15;
    const int half = lane >> 4;          // 0: K pair {0,1}, 1: K pair {2,3}
    const int row  = blockIdx.x * 16 + l;
    const int col  = wave * 16 + l;

    const float* ap = A + (size_t)row * K + half * 2;      // 8B aligned (K%4==0)
    const float* bp = B + (size_t)(half * 2) * N + col;

    v8f acc = {};
    for (int k = 0; k < K; k += 4) {
        v2f a = *(const v2f*)ap;
        ap += 4;
        if (RELU) {
            a.x = fmaxf(a.x, 0.0f);
            a.y = fmaxf(a.y, 0.0f);
        }
        v2f b;
        b.x = bp[0];
        b.y = bp[N];
        bp += (size_t)4 * N;
        acc = __builtin_amdgcn_wmma_f32_16x16x4_f32(false, a, false, b,
                                                    (short)0, acc, false, false);
    }

    float* dp = D + (size_t)(blockIdx.x * 16 + half * 8) * N + wave * 16 + l;
#pragma unroll
    for (int r = 0; r < 8; ++r) dp[(size_t)r * N] = acc[r];
}

// Scalar fallback for leftover rows (M % 16) and odd shapes.
template <bool RELU>
__global__ void gemm_naive_rows(const float* __restrict__ A, const float* __restrict__ B,
                                float* __restrict__ D, int rowStart, int M, int K, int N) {
    int idx = blockIdx.x * blockDim.x + threadIdx.x;
    int row = rowStart + idx / N;
    int col = idx - (idx / N) * N;
    if (row >= M) return;
    float s = 0.0f;
    for (int k = 0; k < K; ++k) {
        float a = A[(size_t)row * K + k];
        if (RELU) a = fmaxf(a, 0.0f);
        s += a * B[(size_t)k * N + col];
    }
    D[(size_t)row * N + col] = s;
}

// ---------------- aggregation ----------------

// out[i,c] = h[i,c] * dinv[i]^2 + bias[c]   (self-loop term; also initializes out)
__global__ void k_selfloop_bias(const float* __restrict__ h, const float* __restrict__ dinv,
                                const float* __restrict__ bias, float* __restrict__ out,
                                int n, int C) {
    int gid = blockIdx.x * blockDim.x + threadIdx.x;
    int i = gid / C;
    if (i >= n) return;
    int c = gid - i * C;
    float di = dinv[i];
    out[gid] = h[gid] * di * di + bias[c];
}

// out[dst,c] += h[src,c] * dinv[src] * dinv[dst]  for each edge, channel-parallel.
__global__ void k_edge_agg(const float* __restrict__ h, const float* __restrict__ dinv,
                           const int* __restrict__ sr, const int* __restrict__ ds,
                           float* __restrict__ out, int E, int C) {
    int gid = blockIdx.x * blockDim.x + threadIdx.x;
    int e = gid / C;
    if (e >= E) return;
    int c = gid - e * C;
    int s = sr[e];
    int d = ds[e];
    float nrm = dinv[s] * dinv[d];
    atomic_add_f32(out + (size_t)d * C + c, h[(size_t)s * C + c] * nrm);
}

// ---------------- host launcher ----------------

static inline int ceil_div(long a, long b) { return (int)((a + b - 1) / b); }

extern "C" void kernel_launch(void* const* d_in, const int* in_sizes, int n_in,
                              void* d_out, int out_size, void* d_ws, size_t ws_size,
                              hipStream_t stream) {
    (void)n_in; (void)out_size; (void)ws_size;

    const float* x  = (const float*)d_in[0];
    const float* W1 = (const float*)d_in[1];
    const float* b1 = (const float*)d_in[2];
    const float* W2 = (const float*)d_in[3];
    const float* b2 = (const float*)d_in[4];
    const int*  src = (const int*)d_in[5];
    const int*  dst = (const int*)d_in[6];
    float*      out = (float*)d_out;

    const int HID = in_sizes[2];           // 128
    const int OUT = in_sizes[4];           // 64
    const int IN  = in_sizes[1] / HID;     // 128
    const int N   = in_sizes[0] / IN;      // 100000
    const int E   = in_sizes[5];           // 600000

    // workspace carve-out (256B aligned)
    char* w = (char*)d_ws;
    auto carve = [&](size_t bytes) -> void* {
        void* p = (void*)w;
        w += (bytes + 255) & ~(size_t)255;
        return p;
    };
    unsigned* deg  = (unsigned*)carve((size_t)N * 4);
    float*    dinv = (float*)carve((size_t)N * 4);
    float*    h1   = (float*)carve((size_t)N * HID * 4);
    float*    a1   = (float*)carve((size_t)N * HID * 4);
    float*    h2   = (float*)carve((size_t)N * OUT * 4);

    // --- degrees / symmetric norm ---
    k_deg_init<<<ceil_div(N, 256), 256, 0, stream>>>(deg, N);
    k_deg_count<<<ceil_div(E, 256), 256, 0, stream>>>(dst, deg, E);
    k_dinv<<<ceil_div(N, 256), 256, 0, stream>>>(deg, dinv, N);

    const int mt  = N / 16;
    const int rem = N - mt * 16;
    const bool wmma_ok = (IN % 4 == 0) && (HID % 4 == 0) && (HID % 16 == 0) && (OUT % 16 == 0);

    // --- layer 1: h1 = x @ W1 ---
    if (wmma_ok && mt > 0) {
        gemm16_wmma<false><<<mt, 32 * (HID / 16), 0, stream>>>(x, W1, h1, IN, HID);
        if (rem)
            gemm_naive_rows<false><<<ceil_div((long)rem * HID, 128), 128, 0, stream>>>(
                x, W1, h1, mt * 16, N, IN, HID);
    } else {
        gemm_naive_rows<false><<<ceil_div((long)N * HID, 128), 128, 0, stream>>>(
            x, W1, h1, 0, N, IN, HID);
    }

    // --- layer 1 aggregation: a1 = D^-1/2 (A+I) D^-1/2 h1 + b1 ---
    k_selfloop_bias<<<ceil_div((long)N * HID, 256), 256, 0, stream>>>(h1, dinv, b1, a1, N, HID);
    k_edge_agg<<<ceil_div((long)E * HID, 256), 256, 0, stream>>>(h1, dinv, src, dst, a1, E, HID);

    // --- layer 2: h2 = relu(a1) @ W2  (ReLU folded into A-load) ---
    if (wmma_ok && mt > 0) {
        gemm16_wmma<true><<<mt, 32 * (OUT / 16), 0, stream>>>(a1, W2, h2, HID, OUT);
        if (rem)
            gemm_naive_rows<true><<<ceil_div((long)rem * OUT, 128), 128, 0, stream>>>(
                a1, W2, h2, mt * 16, N, HID, OUT);
    } else {
        gemm_naive_rows<true><<<ceil_div((long)N * OUT, 128), 128, 0, stream>>>(
            a1, W2, h2, 0, N, HID, OUT);
    }

    // --- layer 2 aggregation straight into d_out ---
    k_selfloop_bias<<<ceil_div((long)N * OUT, 256), 256, 0, stream>>>(h2, dinv, b2, out, N, OUT);
    k_edge_agg<<<ceil_div((long)E * OUT, 256), 256, 0, stream>>>(h2, dinv, src, dst, out, E, OUT);
}